// EfficientAttention_36060545417823
// MI455X (gfx1250) — compile-verified
//
#include <hip/hip_runtime.h>
#include <math.h>

#define BATCH 4
#define HEADS 8
#define SEQ   16384
#define DIM   64
#define BHN   (BATCH*HEADS)          // 32
#define SCL   0.125f                 // 1/sqrt(64)

#define CHA    64                    // Q-stat chunks per (b,h)
#define ROWS_A (SEQ/CHA)             // 256
#define CHB    32                    // ctx chunks per (b,h)
#define ROWS_B (SEQ/CHB)             // 512
#define CHC    32                    // output chunks per (b,h)
#define ROWS_C (SEQ/CHC)             // 512
#define PANEL  64                    // rows staged in LDS per step
#define PAD    4                     // float4-aligned LDS padding (stride 68)
#define PD4    (PANEL*DIM/4)         // 1024 float4 per panel

typedef __attribute__((ext_vector_type(2))) float v2f;
typedef __attribute__((ext_vector_type(8))) float v8f;

// ---- workspace layout (float offsets) ----
#define OFF_PARTA 0
#define SZ_PARTA  (BHN*CHA*DIM*2)                 // 262144 floats
#define OFF_CMAX  (OFF_PARTA + SZ_PARTA)
#define OFF_CINV  (OFF_CMAX + BHN*DIM)
#define OFF_PCTX  (OFF_CINV + BHN*DIM)
#define SZ_PCTX   (BHN*CHB*DIM*DIM)               // 4194304 floats
#define OFF_CTX   (OFF_PCTX + SZ_PCTX)            // + 131072 floats

__device__ __forceinline__ void smax_merge(float& m, float& s, float m2, float s2) {
    float M = fmaxf(m, m2);
    s = s * __expf(m - M) + s2 * __expf(m2 - M);
    m = M;
}
__device__ __forceinline__ void online_update(float& m, float& s, float x) {
    if (x > m) { s = s * __expf(m - x) + 1.0f; m = x; }
    else       { s += __expf(x - m); }
}

// ---------------- Kernel 1: Q column-softmax partial stats (float4 loads) ----------------
__global__ void ea_colstats_partial(const float* __restrict__ Q, float* __restrict__ ws) {
    const int ch = blockIdx.x;                 // 0..CHA-1
    const int bh = blockIdx.y;                 // 0..31
    const int t  = threadIdx.x;                // 0..255
    const int cg = t & 15;                     // column group: cols cg*4..cg*4+3
    const int g  = t >> 4;                     // row group 0..15
    const float4* Q4 = (const float4*)(Q + (bh * SEQ + ch * ROWS_A) * DIM);

    float m0 = -INFINITY, m1 = -INFINITY, m2 = -INFINITY, m3 = -INFINITY;
    float s0 = 0.f, s1 = 0.f, s2 = 0.f, s3 = 0.f;
    for (int i = g; i < ROWS_A; i += 16) {     // 16 iterations, full row per 16 threads
        float4 x = Q4[i * (DIM / 4) + cg];
        online_update(m0, s0, x.x * SCL);
        online_update(m1, s1, x.y * SCL);
        online_update(m2, s2, x.z * SCL);
        online_update(m3, s3, x.w * SCL);
    }
    __shared__ float sm[16][64];
    __shared__ float ss[16][64];
    const int c0 = cg * 4;
    sm[g][c0 + 0] = m0; ss[g][c0 + 0] = s0;
    sm[g][c0 + 1] = m1; ss[g][c0 + 1] = s1;
    sm[g][c0 + 2] = m2; ss[g][c0 + 2] = s2;
    sm[g][c0 + 3] = m3; ss[g][c0 + 3] = s3;
    __syncthreads();
    if (t < 64) {
        float M = sm[0][t], S = ss[0][t];
        #pragma unroll
        for (int j = 1; j < 16; ++j) smax_merge(M, S, sm[j][t], ss[j][t]);
        float* part = ws + OFF_PARTA;
        int idx = ((bh * CHA + ch) * DIM + t) * 2;
        part[idx + 0] = M;
        part[idx + 1] = S;
    }
}

// ---------------- Kernel 2: merge partial stats ----------------
__global__ void ea_colstats_final(float* __restrict__ ws) {
    const int tid = blockIdx.x * 256 + threadIdx.x;   // 0..2047
    const int bh = tid >> 6, d = tid & 63;
    const float* part = ws + OFF_PARTA;
    float M = -INFINITY, S = 0.0f;
    for (int ch = 0; ch < CHA; ++ch) {
        int idx = ((bh * CHA + ch) * DIM + d) * 2;
        smax_merge(M, S, part[idx], part[idx + 1]);
    }
    ws[OFF_CMAX + bh * DIM + d] = M;
    ws[OFF_CINV + bh * DIM + d] = 1.0f / S;
}

// ---------------- Kernel 3: ctx partial = softmax_row(K)^T @ V (WMMA fp32) ----------------
__global__ void ea_ctx_partial(const float* __restrict__ Kg, const float* __restrict__ Vg,
                               float* __restrict__ ws) {
    const int ch = blockIdx.x;                 // 0..CHB-1
    const int bh = blockIdx.y;                 // 0..31
    const int t  = threadIdx.x;                // 0..255 (8 waves)
    const int wave = t >> 5;
    const int lane = t & 31;
    const int lh = lane & 15;                  // low-half index
    const int kq = (lane >> 4) * 2;            // 0 or 2

    __shared__ float Ksm[PANEL][DIM + PAD];
    __shared__ float Vsm[PANEL][DIM + PAD];
    __shared__ float red[PANEL][4];

    const int tile0 = wave * 2;
    const int mi0 = tile0 >> 2, ei0 = tile0 & 3;
    const int mi1 = (tile0 + 1) >> 2, ei1 = (tile0 + 1) & 3;

    v8f acc0 = {}; v8f acc1 = {};

    for (int p = 0; p < ROWS_B / PANEL; ++p) {
        const int base4 = ((bh * SEQ + ch * ROWS_B + p * PANEL) * DIM) >> 2;
        const float4* K4 = (const float4*)Kg;
        const float4* V4 = (const float4*)Vg;
        for (int i = t; i < PD4; i += 256) {   // 4 float4 per thread
            int row = i >> 4, col = (i & 15) * 4;
            *(float4*)&Ksm[row][col] = K4[base4 + i];
            *(float4*)&Vsm[row][col] = V4[base4 + i];
        }
        __syncthreads();

        // row softmax of K panel (64 rows x 64 cols), 4 threads per row
        const int r = t >> 2, q4 = t & 3, c0 = q4 * 16;
        float lm = -INFINITY;
        #pragma unroll
        for (int c = 0; c < 16; ++c) lm = fmaxf(lm, Ksm[r][c0 + c] * SCL);
        red[r][q4] = lm;
        __syncthreads();
        float rm = fmaxf(fmaxf(red[r][0], red[r][1]), fmaxf(red[r][2], red[r][3]));
        float ls = 0.0f;
        #pragma unroll
        for (int c = 0; c < 16; ++c) ls += __expf(Ksm[r][c0 + c] * SCL - rm);
        __syncthreads();
        red[r][q4] = ls;
        __syncthreads();
        float inv = 1.0f / (red[r][0] + red[r][1] + red[r][2] + red[r][3]);
        #pragma unroll
        for (int c = 0; c < 16; ++c)
            Ksm[r][c0 + c] = __expf(Ksm[r][c0 + c] * SCL - rm) * inv;
        __syncthreads();

        // ctx_tile[mi,ei] += sum_n Ksm[n][mi*16+m] * Vsm[n][ei*16+e]
        #pragma unroll
        for (int kk = 0; kk < PANEL / 4; ++kk) {
            const int n0 = kk * 4 + kq;
            v2f a0, b0, a1, b1;
            a0.x = Ksm[n0 + 0][mi0 * 16 + lh];
            a0.y = Ksm[n0 + 1][mi0 * 16 + lh];
            b0.x = Vsm[n0 + 0][ei0 * 16 + lh];
            b0.y = Vsm[n0 + 1][ei0 * 16 + lh];
            acc0 = __builtin_amdgcn_wmma_f32_16x16x4_f32(false, a0, false, b0,
                                                         (short)0, acc0, false, false);
            a1.x = Ksm[n0 + 0][mi1 * 16 + lh];
            a1.y = Ksm[n0 + 1][mi1 * 16 + lh];
            b1.x = Vsm[n0 + 0][ei1 * 16 + lh];
            b1.y = Vsm[n0 + 1][ei1 * 16 + lh];
            acc1 = __builtin_amdgcn_wmma_f32_16x16x4_f32(false, a1, false, b1,
                                                         (short)0, acc1, false, false);
        }
        __syncthreads();
    }

    // restage partial ctx through LDS, then coalesced float4 store
    const int rofs = (lane >> 4) * 8;
    #pragma unroll
    for (int rr = 0; rr < 8; ++rr) {
        Ksm[mi0 * 16 + rr + rofs][ei0 * 16 + lh] = acc0[rr];
        Ksm[mi1 * 16 + rr + rofs][ei1 * 16 + lh] = acc1[rr];
    }
    __syncthreads();
    float4* pctx4 = (float4*)(ws + OFF_PCTX + (bh * CHB + ch) * DIM * DIM);
    for (int i = t; i < DIM * DIM / 4; i += 256) {
        int row = i >> 4, col = (i & 15) * 4;
        pctx4[i] = *(const float4*)&Ksm[row][col];
    }
}

// ---------------- Kernel 4: reduce ctx partials (float4) ----------------
__global__ void ea_ctx_reduce(float* __restrict__ ws) {
    const int bh = blockIdx.x;
    const float4* pctx4 = (const float4*)(ws + OFF_PCTX);
    float4* ctx4 = (float4*)(ws + OFF_CTX);
    const int Q4 = DIM * DIM / 4;              // 1024
    for (int e = threadIdx.x; e < Q4; e += 256) {
        float4 s = {0.f, 0.f, 0.f, 0.f};
        for (int ch = 0; ch < CHB; ++ch) {
            float4 v = pctx4[(bh * CHB + ch) * Q4 + e];
            s.x += v.x; s.y += v.y; s.z += v.z; s.w += v.w;
        }
        ctx4[bh * Q4 + e] = s;
    }
}

// ---------------- Kernel 5: out = softmax_col(Q) @ ctx (WMMA fp32) ----------------
__global__ void ea_apply(const float* __restrict__ Qg, const float* __restrict__ ws,
                         float* __restrict__ out) {
    const int ch = blockIdx.x;                 // 0..CHC-1
    const int bh = blockIdx.y;                 // 0..31
    const int t  = threadIdx.x;
    const int wave = t >> 5;
    const int lane = t & 31;
    const int lh = lane & 15;
    const int kq = (lane >> 4) * 2;

    __shared__ float Qsm[PANEL][DIM + PAD];
    __shared__ float Csm[DIM][DIM + PAD];
    __shared__ float cmax[DIM];
    __shared__ float cinv[DIM];

    if (t < DIM) {
        cmax[t] = ws[OFF_CMAX + bh * DIM + t];
        cinv[t] = ws[OFF_CINV + bh * DIM + t];
    }
    {
        const float4* ctx4 = (const float4*)(ws + OFF_CTX + bh * DIM * DIM);
        for (int i = t; i < DIM * DIM / 4; i += 256) {
            int row = i >> 4, col = (i & 15) * 4;
            *(float4*)&Csm[row][col] = ctx4[i];
        }
    }
    __syncthreads();

    const int tile0 = wave * 2;
    const int mi0 = tile0 >> 2, ei0 = tile0 & 3;
    const int mi1 = (tile0 + 1) >> 2, ei1 = (tile0 + 1) & 3;
    const int rofs = (lane >> 4) * 8;
    const float4* Q4 = (const float4*)Qg;
    float4* out4 = (float4*)out;

    for (int p = 0; p < ROWS_C / PANEL; ++p) {
        const int base4 = ((bh * SEQ + ch * ROWS_C + p * PANEL) * DIM) >> 2;
        for (int i = t; i < PD4; i += 256) {
            int row = i >> 4, col = (i & 15) * 4;
            float4 x = Q4[base4 + i];
            float4 r;
            r.x = __expf(x.x * SCL - cmax[col + 0]) * cinv[col + 0];
            r.y = __expf(x.y * SCL - cmax[col + 1]) * cinv[col + 1];
            r.z = __expf(x.z * SCL - cmax[col + 2]) * cinv[col + 2];
            r.w = __expf(x.w * SCL - cmax[col + 3]) * cinv[col + 3];
            *(float4*)&Qsm[row][col] = r;
        }
        __syncthreads();

        v8f acc0 = {}; v8f acc1 = {};
        #pragma unroll
        for (int kk = 0; kk < DIM / 4; ++kk) {
            const int k0 = kk * 4 + kq;
            v2f a0, b0, a1, b1;
            a0.x = Qsm[mi0 * 16 + lh][k0 + 0];
            a0.y = Qsm[mi0 * 16 + lh][k0 + 1];
            b0.x = Csm[k0 + 0][ei0 * 16 + lh];
            b0.y = Csm[k0 + 1][ei0 * 16 + lh];
            acc0 = __builtin_amdgcn_wmma_f32_16x16x4_f32(false, a0, false, b0,
                                                         (short)0, acc0, false, false);
            a1.x = Qsm[mi1 * 16 + lh][k0 + 0];
            a1.y = Qsm[mi1 * 16 + lh][k0 + 1];
            b1.x = Csm[k0 + 0][ei1 * 16 + lh];
            b1.y = Csm[k0 + 1][ei1 * 16 + lh];
            acc1 = __builtin_amdgcn_wmma_f32_16x16x4_f32(false, a1, false, b1,
                                                         (short)0, acc1, false, false);
        }
        __syncthreads();

        // restage out panel through LDS (Qsm is free now), flush with float4 stores
        #pragma unroll
        for (int rr = 0; rr < 8; ++rr) {
            Qsm[mi0 * 16 + rr + rofs][ei0 * 16 + lh] = acc0[rr];
            Qsm[mi1 * 16 + rr + rofs][ei1 * 16 + lh] = acc1[rr];
        }
        __syncthreads();
        for (int i = t; i < PD4; i += 256) {
            int row = i >> 4, col = (i & 15) * 4;
            out4[base4 + i] = *(const float4*)&Qsm[row][col];
        }
        __syncthreads();
    }
}

extern "C" void kernel_launch(void* const* d_in, const int* in_sizes, int n_in,
                              void* d_out, int out_size, void* d_ws, size_t ws_size,
                              hipStream_t stream) {
    (void)in_sizes; (void)n_in; (void)out_size; (void)ws_size;
    const float* Q = (const float*)d_in[0];
    const float* K = (const float*)d_in[1];
    const float* V = (const float*)d_in[2];
    float* out = (float*)d_out;
    float* ws  = (float*)d_ws;

    ea_colstats_partial<<<dim3(CHA, BHN), 256, 0, stream>>>(Q, ws);
    ea_colstats_final<<<dim3((BHN * DIM) / 256), 256, 0, stream>>>(ws);
    ea_ctx_partial<<<dim3(CHB, BHN), 256, 0, stream>>>(K, V, ws);
    ea_ctx_reduce<<<dim3(BHN), 256, 0, stream>>>(ws);
    ea_apply<<<dim3(CHC, BHN), 256, 0, stream>>>(Q, ws, out);
}